// PCGNN_7645041787293
// MI455X (gfx1250) — compile-verified
//
#include <hip/hip_runtime.h>
#include <hip/hip_bf16.h>
#include <math.h>

typedef __attribute__((ext_vector_type(2))) float v2f;
typedef __attribute__((ext_vector_type(8))) float v8f;

#define TPW 4   // complete 16-edge tiles per wave in the main kernel

// ---------------------------------------------------------------------------
// Workspace layout (floats):
//   P    [256]      : folded params: Weff[4][32] (row 3 = 0), beff[32] @128,
//                     veff[34] @160, ceff @194
//   aggr [N*32]     : per-node accumulated messages
//   cnt  [N]        : per-node in-degree (computed once; reused by all convs)
//   xA   [N*2]      : node features ping
//   xB   [N*2]      : node features pong
// ---------------------------------------------------------------------------

__device__ __forceinline__ void atomAddF(float* p, float v) {
#if defined(__AMDGCN__)
    unsafeAtomicAdd(p, v);
#else
    atomicAdd(p, v);
#endif
}

// Single-instruction ReLU. Plain fmaxf/fmed3 lower to canonicalize+max because
// LLVM can't prove the WMMA result is canonical; inline asm avoids that.
__device__ __forceinline__ float reluf(float v) {
#if defined(__AMDGCN__)
    float r;
    asm("v_max_num_f32 %0, 0, %1" : "=v"(r) : "v"(v));
    return r;
#else
    return fmaxf(v, 0.0f);
#endif
}

// Fold the two 3-layer MLPs (no inner nonlinearities) into single affine maps.
__global__ void pcgnn_fold_params(
    const float* __restrict__ W1, const float* __restrict__ b1,
    const float* __restrict__ W2, const float* __restrict__ b2,
    const float* __restrict__ W3, const float* __restrict__ b3,
    const float* __restrict__ V1, const float* __restrict__ c1,
    const float* __restrict__ V2, const float* __restrict__ c2,
    const float* __restrict__ V3, const float* __restrict__ c3,
    float* __restrict__ P)
{
    __shared__ float W12[3 * 32];   // W1 @ W2
    __shared__ float b12[32];       // b1 @ W2 + b2
    __shared__ float V12[34 * 16];  // V1 @ V2
    __shared__ float c12[16];       // c1 @ V2 + c2
    const int t = threadIdx.x;      // 128 threads

    for (int idx = t; idx < 96; idx += 128) {
        int i = idx >> 5, j = idx & 31;
        float s = 0.f;
        for (int k = 0; k < 32; ++k) s += W1[i * 32 + k] * W2[k * 32 + j];
        W12[idx] = s;
    }
    for (int j = t; j < 32; j += 128) {
        float s = b2[j];
        for (int k = 0; k < 32; ++k) s += b1[k] * W2[k * 32 + j];
        b12[j] = s;
    }
    for (int idx = t; idx < 34 * 16; idx += 128) {
        int i = idx >> 4, j = idx & 15;
        float s = 0.f;
        for (int k = 0; k < 32; ++k) s += V1[i * 32 + k] * V2[k * 16 + j];
        V12[idx] = s;
    }
    for (int j = t; j < 16; j += 128) {
        float s = c2[j];
        for (int k = 0; k < 32; ++k) s += c1[k] * V2[k * 16 + j];
        c12[j] = s;
    }
    __syncthreads();

    // Weff = W12 @ W3  (rows 0..2), row 3 zero-padded for the K=4 WMMA
    for (int idx = t; idx < 128; idx += 128) {
        int i = idx >> 5, j = idx & 31;
        float s = 0.f;
        if (i < 3) for (int k = 0; k < 32; ++k) s += W12[i * 32 + k] * W3[k * 32 + j];
        P[idx] = s;
    }
    for (int j = t; j < 32; j += 128) {          // beff
        float s = b3[j];
        for (int k = 0; k < 32; ++k) s += b12[k] * W3[k * 32 + j];
        P[128 + j] = s;
    }
    for (int i = t; i < 34; i += 128) {          // veff
        float s = 0.f;
        for (int j = 0; j < 16; ++j) s += V12[i * 16 + j] * V3[j];
        P[160 + i] = s;
    }
    if (t == 0) {                                // ceff
        float s = c3[0];
        for (int j = 0; j < 16; ++j) s += c12[j] * V3[j];
        P[194] = s;
    }
}

__global__ void pcgnn_zero(float* __restrict__ p, int n) {
    int i = blockIdx.x * blockDim.x + threadIdx.x;
    if (i < n) p[i] = 0.f;
}

// In-degree, computed ONCE (invariant across the three conv passes).
__global__ void pcgnn_degree(const int* __restrict__ ei, float* __restrict__ cnt,
                             int E) {
    int e = blockIdx.x * blockDim.x + threadIdx.x;
    if (e < E) atomAddF(cnt + (unsigned)ei[E + e], 1.0f);
}

// Main edge kernel: one wave32 per TPW *complete* 16-edge tiles.
//   h[16x32] = relu( msg[16x4] @ Weff[4x32] + beff )  (2x V_WMMA_F32_16X16X4_F32)
// Branch-free body; the only branch is the wave-uniform early-out, so EXEC is
// provably all-ones at every WMMA.
__global__ void pcgnn_edge_wmma(
    const float* __restrict__ x,        // [N,2] current node features
    const float* __restrict__ ea,       // [E,1]
    const int*   __restrict__ ei,       // [2,E] (src row then dst row)
    const float* __restrict__ P,
    float* __restrict__ aggr,           // [N,32]
    int E, int nWaves)
{
    const int wave = (blockIdx.x * blockDim.x + threadIdx.x) >> 5;
    const int lane = threadIdx.x & 31;
    if (wave >= nWaves) return;         // wave-uniform

    const int  nl    = lane & 15;
    const bool hi    = lane >= 16;
    const int  mbase = hi ? 8 : 0;

    // --- B matrix 4x16 f32 (loop-invariant): VGPR0 = K{0|2}, VGPR1 = K{1|3} ---
    v2f b0, b1;
    b0.x = hi ? P[64 + nl]  : P[nl];
    b0.y = hi ? P[96 + nl]  : P[32 + nl];
    b1.x = hi ? P[80 + nl]  : P[16 + nl];
    b1.y = hi ? P[112 + nl] : P[48 + nl];

    // --- C = broadcast folded bias along M (loop-invariant) ---
    const float be0 = P[128 + nl], be1 = P[144 + nl];
    v8f c0, c1;
#pragma unroll
    for (int r = 0; r < 8; ++r) { c0[r] = be0; c1[r] = be1; }

    // --- Loop-invariant bpermute byte indices: lane (mbase+r) ---
    int bidx[8];
#pragma unroll
    for (int r = 0; r < 8; ++r) bidx[r] = (mbase + r) * 4;

    int e0 = wave * (TPW * 16);
#pragma unroll
    for (int i = 0; i < TPW; ++i, e0 += 16) {
        const int eld  = e0 + nl;
        const int srcv = ei[eld];       // all lanes (hi lanes: harmless dup)
        const int dstv = ei[E + eld];   // lane k holds dst of edge e0+k

        // --- Branch-free A load: msg = [x_src0, x_src1, edge_attr, 0] ---
        // lo lanes: (x[2*src], x[2*src+1]); hi lanes: (ea[e], 0)
        const unsigned sx = 2u * (unsigned)srcv;
        const float* pax = hi ? (ea + eld) : (x + sx);
        const float* pay = hi ? (ea + eld) : (x + sx + 1);
        v2f a;
        a.x = *pax;
        const float ay = *pay;
        a.y = hi ? 0.f : ay;

        v8f d0 = __builtin_amdgcn_wmma_f32_16x16x4_f32(false, a, false, b0,
                                                       (short)0, c0, false, false);
        v8f d1 = __builtin_amdgcn_wmma_f32_16x16x4_f32(false, a, false, b1,
                                                       (short)0, c1, false, false);

        // --- ReLU + scatter: D row r holds M = r + 8*hi, col = nl (+16) ---
#pragma unroll
        for (int r = 0; r < 8; ++r) {
            const int d = __builtin_amdgcn_ds_bpermute(bidx[r], dstv);
            const unsigned base = (unsigned)d * 32u + (unsigned)nl;  // 32-bit: saddr form
            atomAddF(aggr + base,      reluf(d0[r]));
            atomAddF(aggr + base + 16, reluf(d1[r]));
        }
    }
}

// Cleanup for edges not covered by complete TPW-tile waves (< TPW*16+15 edges).
// One wave32 per edge; lane n computes output column n.
__global__ void pcgnn_edge_tail(
    const float* __restrict__ x, const float* __restrict__ ea,
    const int* __restrict__ ei, const float* __restrict__ P,
    float* __restrict__ aggr, int E, int eStart)
{
    const int e    = eStart + ((blockIdx.x * blockDim.x + threadIdx.x) >> 5);
    const int lane = threadIdx.x & 31;
    if (e >= E) return;
    const int src = ei[e], dst = ei[E + e];
    const float x0 = x[2u * (unsigned)src], x1 = x[2u * (unsigned)src + 1];
    const float ev = ea[e];
    float h = x0 * P[lane] + x1 * P[32 + lane] + ev * P[64 + lane] + P[128 + lane];
    atomAddF(aggr + ((unsigned)dst * 32u + (unsigned)lane), reluf(h));
}

// One wave32 per node: comb = sigmoid(x·veff[0:2] + mean(aggr)·veff[2:34] + ceff)
__global__ void pcgnn_node(
    const float* __restrict__ x,
    const float* __restrict__ aggr,
    const float* __restrict__ cnt,
    const float* __restrict__ P,
    float* __restrict__ xn,
    int N)
{
    const int n    = (blockIdx.x * blockDim.x + threadIdx.x) >> 5;
    const int lane = threadIdx.x & 31;
    if (n >= N) return;

    float p = aggr[(unsigned)n * 32u + (unsigned)lane] * P[162 + lane];
#pragma unroll
    for (int off = 16; off > 0; off >>= 1) p += __shfl_xor(p, off, 32);

    if (lane == 0) {
        const float c  = fmaxf(cnt[n], 1.0f);
        const float x0 = x[2u * (unsigned)n];
        const float x1 = x[2u * (unsigned)n + 1];
        const float s  = p / c + x0 * P[160] + x1 * P[161] + P[194];
        const float comb = 1.0f / (1.0f + __expf(-s));
        xn[2u * (unsigned)n]     = comb;
        xn[2u * (unsigned)n + 1] = x1;
    }
}

extern "C" void kernel_launch(void* const* d_in, const int* in_sizes, int n_in,
                              void* d_out, int out_size, void* d_ws, size_t ws_size,
                              hipStream_t stream) {
    const float* x  = (const float*)d_in[0];
    const float* ea = (const float*)d_in[1];
    const int*   ei = (const int*)d_in[2];   // JAX x64-off: "int64" -> int32
    const float* W1 = (const float*)d_in[3];
    const float* b1 = (const float*)d_in[4];
    const float* W2 = (const float*)d_in[5];
    const float* b2 = (const float*)d_in[6];
    const float* W3 = (const float*)d_in[7];
    const float* b3 = (const float*)d_in[8];
    const float* V1 = (const float*)d_in[9];
    const float* c1 = (const float*)d_in[10];
    const float* V2 = (const float*)d_in[11];
    const float* c2 = (const float*)d_in[12];
    const float* V3 = (const float*)d_in[13];
    const float* c3 = (const float*)d_in[14];

    const int N = in_sizes[0] / 2;
    const int E = in_sizes[1];

    float* ws   = (float*)d_ws;
    float* P    = ws;
    float* aggr = ws + 256;
    float* cnt  = aggr + (size_t)N * 32;
    float* xA   = cnt + N;
    float* xB   = xA + (size_t)N * 2;

    pcgnn_fold_params<<<1, 128, 0, stream>>>(W1, b1, W2, b2, W3, b3,
                                             V1, c1, V2, c2, V3, c3, P);

    // Zero aggr+cnt once, then count in-degree once (invariant across convs).
    pcgnn_zero<<<(N * 33 + 255) / 256, 256, 0, stream>>>(aggr, N * 33);
    pcgnn_degree<<<(E + 255) / 256, 256, 0, stream>>>(ei, cnt, E);

    const int nFullTiles = E / 16;
    const int nWaves     = nFullTiles / TPW;         // complete TPW-tile waves
    const int eMain      = nWaves * TPW * 16;        // edges covered by main
    const int nTailEdges = E - eMain;                // < TPW*16 + 15
    const int edgeBlocks = (nWaves * 32 + 255) / 256;
    const int tailBlocks = (nTailEdges * 32 + 255) / 256;
    const int nodeBlocks = (N * 32 + 255) / 256;
    const int zeroBlocks = (N * 32 + 255) / 256;

    const float* xc = x;
    float* outs[3] = { xA, xB, (float*)d_out };
    for (int it = 0; it < 3; ++it) {
        if (it > 0)
            pcgnn_zero<<<zeroBlocks, 256, 0, stream>>>(aggr, N * 32);
        if (nWaves > 0)
            pcgnn_edge_wmma<<<edgeBlocks, 256, 0, stream>>>(xc, ea, ei, P,
                                                            aggr, E, nWaves);
        if (nTailEdges > 0)
            pcgnn_edge_tail<<<tailBlocks, 256, 0, stream>>>(xc, ea, ei, P,
                                                            aggr, E, eMain);
        pcgnn_node<<<nodeBlocks, 256, 0, stream>>>(xc, aggr, cnt, P, outs[it], N);
        xc = outs[it];
    }
}